// HeteroGraphSAGE_85469849190597
// MI455X (gfx1250) — compile-verified
//
#include <hip/hip_runtime.h>
#include <hip/hip_bf16.h>

typedef float v2f __attribute__((ext_vector_type(2)));
typedef float v8f __attribute__((ext_vector_type(8)));

#define KDIM 128

// ---------------- utility kernels ----------------

__global__ void zero_kernel(float* __restrict__ p, size_t n) {
    size_t i = (size_t)blockIdx.x * blockDim.x + threadIdx.x;
    size_t stride = (size_t)gridDim.x * blockDim.x;
    for (; i < n; i += stride) p[i] = 0.0f;
}

// out = alpha*a + beta*b  (b may be null)
__global__ void axpby_kernel(float* __restrict__ out, const float* __restrict__ a,
                             const float* __restrict__ b, float alpha, float beta, int n) {
    int i = blockIdx.x * blockDim.x + threadIdx.x;
    int stride = gridDim.x * blockDim.x;
    for (; i < n; i += stride)
        out[i] = alpha * a[i] + (b ? beta * b[i] : 0.0f);
}

__global__ void degree_kernel(const int* __restrict__ dst, float* __restrict__ cnt, int nE) {
    int e = blockIdx.x * blockDim.x + threadIdx.x;
    if (e < nE) atomicAdd(&cnt[dst[e]], 1.0f);
}

// one wave per edge: gather 128-float row of x[src], atomic-add into agg[dst]
__global__ void scatter_kernel(const float* __restrict__ x, const int* __restrict__ src,
                               const int* __restrict__ dst, float* __restrict__ agg, int nE) {
    int gt = blockIdx.x * blockDim.x + threadIdx.x;
    int e = gt >> 5;
    int lane = threadIdx.x & 31;
    if (e >= nE) return;
    int s = src[e];
    int d = dst[e];
    const float4 v = ((const float4*)(x + (size_t)s * KDIM))[lane];
    float* o = agg + (size_t)d * KDIM + lane * 4;
    atomicAdd(o + 0, v.x);
    atomicAdd(o + 1, v.y);
    atomicAdd(o + 2, v.z);
    atomicAdd(o + 3, v.w);
}

// ---------------- fused 3-source WMMA GEMM ----------------
// Accumulate one K=128 source into the two 16x16 accumulators.
// Fully unrolled; pointers flow straight from kernel args so the backend can
// prove global address space (global_load_b64, LOADcnt only).
template <bool DO_SCALE>
__device__ __forceinline__ void wmma_acc_src(const float* __restrict__ A,
                                             const float* __restrict__ W,
                                             int rowA0, int rowA1, int colW, int halfK,
                                             float sc0, float sc1, v8f& c0, v8f& c1) {
    const float2* __restrict__ a0p = (const float2*)(A + (size_t)rowA0 * KDIM);
    const float2* __restrict__ a1p = (const float2*)(A + (size_t)rowA1 * KDIM);
    const float2* __restrict__ wp  = (const float2*)(W + (size_t)colW * KDIM);
#pragma unroll
    for (int k = 0; k < KDIM; k += 4) {
        const int fi = (k + halfK) >> 1;   // float2 index within the row
        float2 bf = wp[fi];
        float2 af0 = a0p[fi];
        float2 af1 = a1p[fi];
        v2f bvv; bvv[0] = bf.x; bvv[1] = bf.y;
        v2f av0, av1;
        if (DO_SCALE) {
            av0[0] = af0.x * sc0; av0[1] = af0.y * sc0;
            av1[0] = af1.x * sc1; av1[1] = af1.y * sc1;
        } else {
            av0[0] = af0.x; av0[1] = af0.y;
            av1[0] = af1.x; av1[1] = af1.y;
        }
        c0 = __builtin_amdgcn_wmma_f32_16x16x4_f32(false, av0, false, bvv,
                                                   (short)0, c0, false, false);
        c1 = __builtin_amdgcn_wmma_f32_16x16x4_f32(false, av1, false, bvv,
                                                   (short)0, c1, false, false);
    }
}

// out[M,N] = act( (A0/cnt0)@W0^T + (A1/cnt1)@W1^T + A2@W2^T + bias ),  K = 128
// One wave computes a 32x16 tile of out via V_WMMA_F32_16X16X4_F32.
__global__ __launch_bounds__(256) void sage_gemm_kernel(
    const float* __restrict__ A0, const float* __restrict__ cnt0,
    const float* __restrict__ A1, const float* __restrict__ cnt1,
    const float* __restrict__ A2,
    const float* __restrict__ W0, const float* __restrict__ W1, const float* __restrict__ W2,
    const float* __restrict__ bias, float* __restrict__ out,
    int M, int N, int relu) {
    const int gt = blockIdx.x * blockDim.x + threadIdx.x;
    const int wave = gt >> 5;
    const int lane = threadIdx.x & 31;
    const int nTilesN = N >> 4;
    const int totalTiles = (M >> 5) * nTilesN;   // M is a multiple of 32 here
    if (wave >= totalTiles) return;              // wave-uniform: EXEC all-1s inside

    const int m0 = (wave / nTilesN) << 5;
    const int n0 = (wave % nTilesN) << 4;
    const int r = lane & 15;
    const int halfK = (lane >> 4) << 1;          // K sub-offset 0 or 2 (A/B 16x4 layout)
    const int rowA0 = m0 + r;                    // rows m0..m0+15 (accumulator c0)
    const int rowA1 = m0 + 16 + r;               // rows m0+16..m0+31 (accumulator c1)
    const int colW = n0 + r;                     // C/D column for this lane

    const float bv = bias[colW];
    v8f c0, c1;
#pragma unroll
    for (int i = 0; i < 8; i++) { c0[i] = bv; c1[i] = bv; }

    const float sc00 = 1.0f / fmaxf(cnt0[rowA0], 1.0f);
    const float sc01 = 1.0f / fmaxf(cnt0[rowA1], 1.0f);
    const float sc10 = 1.0f / fmaxf(cnt1[rowA0], 1.0f);
    const float sc11 = 1.0f / fmaxf(cnt1[rowA1], 1.0f);

    wmma_acc_src<true >(A0, W0, rowA0, rowA1, colW, halfK, sc00, sc01, c0, c1);
    wmma_acc_src<true >(A1, W1, rowA0, rowA1, colW, halfK, sc10, sc11, c0, c1);
    wmma_acc_src<false>(A2, W2, rowA0, rowA1, colW, halfK, 1.0f, 1.0f, c0, c1);

    const int halfM = (lane >> 4) << 3;          // 0 or 8 (C/D layout)
#pragma unroll
    for (int i = 0; i < 8; i++) {
        float v0 = c0[i], v1 = c1[i];
        if (relu) { v0 = fmaxf(v0, 0.0f); v1 = fmaxf(v1, 0.0f); }
        out[(size_t)(m0 + halfM + i) * N + colW] = v0;
        out[(size_t)(m0 + 16 + halfM + i) * N + colW] = v1;
    }
}

// ---------------- launch ----------------

extern "C" void kernel_launch(void* const* d_in, const int* in_sizes, int n_in,
                              void* d_out, int out_size, void* d_ws, size_t ws_size,
                              hipStream_t stream) {
    const float* x     = (const float*)d_in[0];
    const int*   ei_s  = (const int*)d_in[1];
    const int*   ei_b  = (const int*)d_in[2];
    const float* W1s_l = (const float*)d_in[3];
    const float* b1s   = (const float*)d_in[4];
    const float* W1s_r = (const float*)d_in[5];
    const float* W1b_l = (const float*)d_in[6];
    const float* b1b   = (const float*)d_in[7];
    const float* W1b_r = (const float*)d_in[8];
    const float* W2s_l = (const float*)d_in[9];
    const float* b2s   = (const float*)d_in[10];
    const float* W2s_r = (const float*)d_in[11];
    const float* W2b_l = (const float*)d_in[12];
    const float* b2b   = (const float*)d_in[13];
    const float* W2b_r = (const float*)d_in[14];

    const int M   = in_sizes[0] / KDIM;   // 100000
    const int E_s = in_sizes[1] / 2;      // 600000
    const int E_b = in_sizes[2] / 2;
    const int* src_s = ei_s;
    const int* dst_s = ei_s + E_s;
    const int* src_b = ei_b;
    const int* dst_b = ei_b + E_b;

    // workspace carve (floats)
    float* w = (float*)d_ws;
    const size_t NM = (size_t)M * KDIM;
    float* aggS = w; w += NM;
    float* aggB = w; w += NM;
    float* h    = w; w += NM;
    float* cntS = w; w += M;
    float* cntB = w; w += M;
    float* Wl1s = w; w += 128 * 128;
    float* Wl1b = w; w += 128 * 128;
    float* Wr1  = w; w += 128 * 128;
    float* bb1  = w; w += 128;
    float* Wl2s = w; w += 64 * 128;
    float* Wl2b = w; w += 64 * 128;
    float* Wr2  = w; w += 64 * 128;
    float* bb2  = w; w += 64;

    const int T = 256;
    auto gsblocks = [](size_t n) { size_t b = (n + 255) / 256; return (int)(b > 8192 ? 8192 : b); };

    // weight prep: fold 0.5 combine into weights/biases
    axpby_kernel<<<64, T, 0, stream>>>(Wl1s, W1s_l, nullptr, 0.5f, 0.0f, 128 * 128);
    axpby_kernel<<<64, T, 0, stream>>>(Wl1b, W1b_l, nullptr, 0.5f, 0.0f, 128 * 128);
    axpby_kernel<<<64, T, 0, stream>>>(Wr1,  W1s_r, W1b_r,   0.5f, 0.5f, 128 * 128);
    axpby_kernel<<<1,  T, 0, stream>>>(bb1,  b1s,   b1b,     0.5f, 0.5f, 128);
    axpby_kernel<<<32, T, 0, stream>>>(Wl2s, W2s_l, nullptr, 0.5f, 0.0f, 64 * 128);
    axpby_kernel<<<32, T, 0, stream>>>(Wl2b, W2b_l, nullptr, 0.5f, 0.0f, 64 * 128);
    axpby_kernel<<<32, T, 0, stream>>>(Wr2,  W2s_r, W2b_r,   0.5f, 0.5f, 64 * 128);
    axpby_kernel<<<1,  T, 0, stream>>>(bb2,  b2s,   b2b,     0.5f, 0.5f, 64);

    // zero accumulators (aggS|aggB contiguous; cntS|cntB contiguous)
    zero_kernel<<<gsblocks(2 * NM), T, 0, stream>>>(aggS, 2 * NM);
    zero_kernel<<<gsblocks(2 * (size_t)M), T, 0, stream>>>(cntS, 2 * (size_t)M);

    // degrees (reused by both layers)
    degree_kernel<<<(E_s + T - 1) / T, T, 0, stream>>>(dst_s, cntS, E_s);
    degree_kernel<<<(E_b + T - 1) / T, T, 0, stream>>>(dst_b, cntB, E_b);

    // layer-1 scatter (segment sums)
    scatter_kernel<<<(int)(((size_t)E_s * 32 + T - 1) / T), T, 0, stream>>>(x, src_s, dst_s, aggS, E_s);
    scatter_kernel<<<(int)(((size_t)E_b * 32 + T - 1) / T), T, 0, stream>>>(x, src_b, dst_b, aggB, E_b);

    // layer-1 fused GEMM + relu: h = relu(aggS/c·Wl1s^T + aggB/c·Wl1b^T + x·Wr1^T + b1)
    {
        int tiles = (M >> 5) * (128 >> 4);
        int blocks = (int)(((size_t)tiles * 32 + T - 1) / T);
        sage_gemm_kernel<<<blocks, T, 0, stream>>>(aggS, cntS, aggB, cntB, x,
                                                   Wl1s, Wl1b, Wr1, bb1, h, M, 128, 1);
    }

    // layer-2 scatter on h
    zero_kernel<<<gsblocks(2 * NM), T, 0, stream>>>(aggS, 2 * NM);
    scatter_kernel<<<(int)(((size_t)E_s * 32 + T - 1) / T), T, 0, stream>>>(h, src_s, dst_s, aggS, E_s);
    scatter_kernel<<<(int)(((size_t)E_b * 32 + T - 1) / T), T, 0, stream>>>(h, src_b, dst_b, aggB, E_b);

    // layer-2 fused GEMM (no relu): out = aggS/c·Wl2s^T + aggB/c·Wl2b^T + h·Wr2^T + b2
    {
        int tiles = (M >> 5) * (64 >> 4);
        int blocks = (int)(((size_t)tiles * 32 + T - 1) / T);
        sage_gemm_kernel<<<blocks, T, 0, stream>>>(aggS, cntS, aggB, cntB, h,
                                                   Wl2s, Wl2b, Wr2, bb2, (float*)d_out, M, 64, 0);
    }
}